// GNCA_63118839382709
// MI455X (gfx1250) — compile-verified
//
#include <hip/hip_runtime.h>

// ---------------------------------------------------------------------------
// GNCA step for MI455X (gfx1250, wave32).
//
//  * prep kernel: nodeH[n][16] = x[n] @ W1x + b1, plus SoA posx/posy/cell
//    arrays in d_ws (~620 KB, L2-resident).
//  * main kernel: 1 block = 8 waves = 8 target nodes, all in the SAME batch.
//    The block stages its batch's full working set into LDS once
//    (nodeH 64 KB + pos/cell 12 KB = 76 KB; WGP has 320 KB), then each wave
//    sweeps the 1024 j's in tiles of 16 entirely out of LDS.
//    Edge MLP 4->16 via V_WMMA_F32_16X16X4_F32 (A: 16 edges x 4 features,
//    B: W1e 4x16, D: 16 edges x 16 hidden).
//    Adjacency per tile captured with ballot_w32 into an SGPR mask:
//      - deg / cell-neighbor counts = popcount of masks (uniform ints)
//      - empty tiles (mask==0) skip WMMA + unpack via scalar branch
//        (EXEC stays all-ones at the WMMA, as the ISA requires)
//      - per-row LDS loads skipped when both half-wave mask bits are 0
//    Self-edge is adjacent in the reference (sqrt(+1e-9)=3.16e-5 > 1e-6),
//    reproduced exactly here, so every wave executes >=1 WMMA.
// ---------------------------------------------------------------------------

typedef __attribute__((ext_vector_type(2))) float v2f;
typedef __attribute__((ext_vector_type(8))) float v8f;

#define GNCA_B 8
#define GNCA_N 1024
#define GNCA_BN (GNCA_B * GNCA_N)
#define GNCA_C 10
#define GNCA_H 16
#define GNCA_O 7

// ---- prep: nodeH = x @ W1[4:14] + b1 ; SoA pos/cell ------------------------
__global__ void gnca_prep(const float* __restrict__ x,
                          const float* __restrict__ W1,
                          const float* __restrict__ b1,
                          float* __restrict__ nodeH,
                          float* __restrict__ posx,
                          float* __restrict__ posy,
                          float* __restrict__ cellr) {
  int nidx = blockIdx.x * blockDim.x + threadIdx.x;
  if (nidx >= GNCA_BN) return;
  const float* xr = x + nidx * GNCA_C;
  float xi[GNCA_C];
#pragma unroll
  for (int c = 0; c < GNCA_C; ++c) xi[c] = xr[c];
  posx[nidx] = xi[0];
  posy[nidx] = xi[1];
  cellr[nidx] = xi[4];
#pragma unroll
  for (int h = 0; h < GNCA_H; ++h) {
    float acc = b1[h];
#pragma unroll
    for (int c = 0; c < GNCA_C; ++c)
      acc = fmaf(xi[c], W1[(4 + c) * GNCA_H + h], acc);
    nodeH[nidx * GNCA_H + h] = acc;
  }
}

// ---- main: one wave32 per target node; batch working set staged in LDS ----
__global__ void __launch_bounds__(256)
gnca_main(const float* __restrict__ x,
          const float* __restrict__ W1,
          const float* __restrict__ W2,
          const float* __restrict__ b2,
          const float* __restrict__ nodeH,
          const float* __restrict__ posx,
          const float* __restrict__ posy,
          const float* __restrict__ cellr,
          float* __restrict__ out) {
  __shared__ float sH[GNCA_N * GNCA_H];   // 64 KB: per-batch node terms
  __shared__ float spx[GNCA_N];           // 4 KB
  __shared__ float spy[GNCA_N];           // 4 KB
  __shared__ float scl[GNCA_N];           // 4 KB

  const int lane = threadIdx.x & 31;
  const int wib  = threadIdx.x >> 5;
  const int gw   = blockIdx.x * (blockDim.x >> 5) + wib;   // 0 .. 8191
  const int b    = gw >> 10;               // same for all 8 waves of a block
  const int i    = gw & 1023;
  const int bn   = b << 10;

  // ---- stage batch working set into LDS (vectorized b128 where possible) --
  {
    const float4* gH4 = (const float4*)(nodeH + bn * GNCA_H);
    float4*       sH4 = (float4*)sH;
#pragma unroll 4
    for (int t = threadIdx.x; t < GNCA_N * GNCA_H / 4; t += 256)
      sH4[t] = gH4[t];
    for (int t = threadIdx.x; t < GNCA_N; t += 256) {
      spx[t] = posx[bn + t];
      spy[t] = posy[bn + t];
      scl[t] = cellr[bn + t];
    }
  }
  __syncthreads();

  const bool hi = (lane >= 16);
  const int  n  = lane & 15;                 // hidden-unit / A-row index

  // target-node position (uniform across wave)
  const float pix = spx[i];
  const float piy = spy[i];

  // B matrix (W1e = W1[0:4][0:16]) in wmma layout: K = (hi?2:0)+{0,1}, N = n
  const int k0 = hi ? 2 : 0;
  v2f bm;
  bm.x = W1[k0 * GNCA_H + n];
  bm.y = W1[(k0 + 1) * GNCA_H + n];

  float agg  = 0.0f;  // agg_h[i][n] per lane; halves combined at the end
  int   degi = 0;     // degree (uniform)
  int   cnbi = 0;     // cell-neighbor count (uniform)

  for (int jt = 0; jt < GNCA_N / 16; ++jt) {
    const int base = jt * 16;
    const int j    = base + n;               // both half-waves: same j
    const float pjx = spx[j];
    const float pjy = spy[j];
    const float tj  = scl[j];
    const float dx  = pjx - pix;             // rel = pos_j - pos_i
    const float dy  = pjy - piy;
    const float dist = sqrtf(dx * dx + dy * dy + 1e-9f);
    const bool  adj  = (dist < 0.04f) && (dist > 1e-6f);

    // wave-uniform adjacency mask for this tile (bits 16..31 duplicate 0..15)
    const unsigned amask = __builtin_amdgcn_ballot_w32(adj);
    const unsigned cmask = __builtin_amdgcn_ballot_w32(adj && (tj > 0.5f));
    degi += __popc(amask & 0xFFFFu);
    cnbi += __popc(cmask & 0xFFFFu);

    if (amask != 0u) {                       // scalar branch; EXEC unchanged
      // A matrix: row M = n (edge i -> base+n), K features split by half-wave
      v2f am;
      am.x = hi ? dy : dist;                 // K=2 : K=0
      am.y = hi ? tj : dx;                   // K=3 : K=1

      v8f c = {0.f, 0.f, 0.f, 0.f, 0.f, 0.f, 0.f, 0.f};
      v8f d = __builtin_amdgcn_wmma_f32_16x16x4_f32(
          /*neg_a=*/false, am, /*neg_b=*/false, bm,
          /*c_mod=*/(short)0, c, /*reuse_a=*/false, /*reuse_b=*/false);

      // unpack: D VGPR v holds edge row M = v + 8*hi, hidden unit N = n
#pragma unroll
      for (int v = 0; v < 8; ++v) {
        // uniform skip when neither half-wave row is adjacent
        if ((((amask >> v) | (amask >> (v + 8))) & 1u) != 0u) {
          const int jl = v + (hi ? 8 : 0);
          const float am2 = ((amask >> jl) & 1u) ? 1.0f : 0.0f;
          const float nh  = sH[(base + jl) * GNCA_H + n];
          float hv = d[v] + nh;              // edge + node + b1
          hv = hv > 0.0f ? hv : 0.0f;        // relu
          agg = fmaf(am2, hv, agg);
        }
      }
    }
  }

  // combine the two half-wave partial sums of agg_h (same hidden unit n)
  agg += __shfl_xor(agg, 16, 32);

  // out = agg_h @ W2  (computed uniformly on all lanes via broadcasts)
  float ov[GNCA_O];
#pragma unroll
  for (int o = 0; o < GNCA_O; ++o) ov[o] = 0.0f;
  for (int h = 0; h < GNCA_H; ++h) {
    const float av = __shfl(agg, h, 32);
#pragma unroll
    for (int o = 0; o < GNCA_O; ++o)
      ov[o] = fmaf(av, W2[h * GNCA_O + o], ov[o]);
  }

  const float deg = (float)degi;

  // ---- per-node state update (uniform; lane 0 stores) ----------------------
  const float* xr = x + (bn + i) * GNCA_C;
  const float p0 = xr[0], p1 = xr[1];
  const float v0 = xr[2], v1 = xr[3];
  const float c4 = xr[4];
  const bool  isc = (c4 > 0.5f);
  const float cf  = isc ? 1.0f : 0.0f;

  const float o0 = (ov[0] + deg * b2[0]) * cf;
  const float o1 = (ov[1] + deg * b2[1]) * cf;

  const float a0 = o0 * 0.02f;               // ACCEL_SCALE
  const float a1 = o1 * 0.02f;
  float nv0 = fminf(fmaxf(v0 + a0, -0.02f), 0.02f);
  float nv1 = fminf(fmaxf(v1 + a1, -0.02f), 0.02f);
  nv0 = isc ? nv0 : v0;                      // where(is_cell, new_vel, vel)
  nv1 = isc ? nv1 : v1;
  float np0 = fminf(fmaxf(p0 + nv0 * cf, -1.0f), 1.0f);
  float np1 = fminf(fmaxf(p1 + nv1 * cf, -1.0f), 1.0f);
  np0 = isc ? np0 : p0;
  np1 = isc ? np1 : p1;

  const bool dead = isc && (degi < 3);       // EDGES_TO_STAY_ALIVE
  const bool cons = (!isc) && (cnbi >= 1);   // CONSUME_EDGES_REQ
  const float kf  = (dead || cons) ? 0.0f : 1.0f;

  if (lane == 0) {
    float* orow = out + (bn + i) * GNCA_C;
    orow[0] = np0 * kf;
    orow[1] = np1 * kf;
    orow[2] = nv0 * kf;
    orow[3] = nv1 * kf;
#pragma unroll
    for (int c = 4; c < GNCA_C; ++c) orow[c] = xr[c] * kf;
  }
}

extern "C" void kernel_launch(void* const* d_in, const int* in_sizes, int n_in,
                              void* d_out, int out_size, void* d_ws, size_t ws_size,
                              hipStream_t stream) {
  const float* x  = (const float*)d_in[0];   // (8,1024,10)
  const float* W1 = (const float*)d_in[1];   // (14,16)
  const float* b1 = (const float*)d_in[2];   // (16,)
  const float* W2 = (const float*)d_in[3];   // (16,7)
  const float* b2 = (const float*)d_in[4];   // (7,)
  float* out = (float*)d_out;

  float* nodeH = (float*)d_ws;                       // 8192*16 floats
  float* posx  = nodeH + GNCA_BN * GNCA_H;           // 8192
  float* posy  = posx + GNCA_BN;                     // 8192
  float* cellr = posy + GNCA_BN;                     // 8192

  gnca_prep<<<dim3((GNCA_BN + 255) / 256), dim3(256), 0, stream>>>(
      x, W1, b1, nodeH, posx, posy, cellr);

  // one wave32 per node: 8192 waves = 1024 blocks * 8 waves
  gnca_main<<<dim3(GNCA_BN / 8), dim3(256), 0, stream>>>(
      x, W1, W2, b2, nodeH, posx, posy, cellr, out);
}